// BlurNet_5789615915109
// MI455X (gfx1250) — compile-verified
//
#include <hip/hip_runtime.h>
#include <cstdint>

// Problem dims (fixed by the reference)
#define IMG_H 720
#define IMG_W 1280
#define IMG_B 16

typedef __attribute__((ext_vector_type(4))) float v4f;  // true clang vector: NT-builtin OK

constexpr int TILE = 64;            // 64x64 output tile per block
constexpr int HALO = 8;             // covers |offset| <= 7 (P(miss) ~ 2.6e-12/tap)
constexpr int STW  = TILE + 2*HALO; // 80 staged cols (row = 320B, 16B aligned)
constexpr int STH  = TILE + 2*HALO; // 80 staged rows
constexpr int NGRP = STH * (STW/4); // 1600 16-byte groups to stage

__device__ __forceinline__ float tap_global(const float* __restrict__ img, int y, int x) {
    return (y >= 0 && y < IMG_H && x >= 0 && x < IMG_W) ? img[y * IMG_W + x] : 0.0f;
}

__device__ __forceinline__ float sample_px(const float* __restrict__ smem,
                                           const float* __restrict__ img,
                                           int y, int x, float dy, float dx,
                                           int gy0, int gx0)
{
    const float py = (float)y + dy;
    const float px = (float)x + dx;
    const float fy0 = floorf(py);
    const float fx0 = floorf(px);
    const int   y0  = (int)fy0;
    const int   x0  = (int)fx0;
    const float wy1 = py - fy0, wx1 = px - fx0;
    const float wy0 = 1.0f - wy1, wx0 = 1.0f - wx1;

    const int ly  = y0 - gy0;              // staged-local coords
    const int lxs = x0 - gx0;
    float v00, v01, v10, v11;
    if (ly >= 0 && ly < STH - 1 && lxs >= 0 && lxs < STW - 1) {
        const float* s = &smem[ly * STW + lxs];
        v00 = s[0];       v01 = s[1];
        v10 = s[STW];     v11 = s[STW + 1];
    } else {                               // statistically-never outlier path
        v00 = tap_global(img, y0,     x0);
        v01 = tap_global(img, y0,     x0 + 1);
        v10 = tap_global(img, y0 + 1, x0);
        v11 = tap_global(img, y0 + 1, x0 + 1);
    }
    return wy0 * (wx0 * v00 + wx1 * v01) + wy1 * (wx0 * v10 + wx1 * v11);
}

__global__ __launch_bounds__(256)
void blurnet_warp_kernel(const float* __restrict__ realB,
                         const float* __restrict__ offset,
                         const float* __restrict__ weight,
                         const float* __restrict__ bias,
                         float* __restrict__ out)
{
    __shared__ __align__(16) float smem[STH * STW];  // 25.6 KB

    const int tx0 = blockIdx.x * TILE;
    const int ty0 = blockIdx.y * TILE;
    const int b   = blockIdx.z;
    const int tid = threadIdx.x;

    const size_t planeHW = (size_t)IMG_H * IMG_W;
    const float* img  = realB  + (size_t)b * planeHW;        // uniform -> SGPR pair
    const float* offY = offset + (size_t)(2 * b) * planeHW;  // plane 0
    const float* offX = offY + planeHW;                      // plane 1
    float* outp = out + (size_t)b * planeHW;

    const int gy0 = ty0 - HALO;
    const int gx0 = tx0 - HALO;

    // ---- Stage 80x80 source tile (zero-padded) into LDS ------------------
    // Interior groups: async global->LDS B128 copies (ASYNCcnt path, RT hint
    // so the source image stays L2-resident). Edge groups: DS-store fill.
    for (int g = tid; g < NGRP; g += 256) {
        const int sy  = g / (STW / 4);
        const int sx4 = (g - sy * (STW / 4)) * 4;
        const int gy  = gy0 + sy;
        const int gx  = gx0 + sx4;
        float* dst = &smem[sy * STW + sx4];
        if (gy >= 0 && gy < IMG_H && gx >= 0 && gx + 3 < IMG_W) {
            const unsigned lds_addr = (unsigned)(uintptr_t)dst;   // low 32b = LDS byte addr
            const int goff = (gy * IMG_W + gx) * 4;               // 16B-aligned byte offset
            asm volatile("global_load_async_to_lds_b128 %0, %1, %2"
                         :: "v"(lds_addr), "v"(goff), "s"(img)
                         : "memory");
        } else {
            dst[0] = tap_global(img, gy, gx + 0);
            dst[1] = tap_global(img, gy, gx + 1);
            dst[2] = tap_global(img, gy, gx + 2);
            dst[3] = tap_global(img, gy, gx + 3);
        }
    }
    asm volatile("s_wait_asynccnt 0" ::: "memory");  // drain this wave's async copies
    __syncthreads();                                 // all waves' staging visible

    const float wgt = weight[0];
    const float bia = bias[0];

    // ---- Gather + bilinear blend from LDS --------------------------------
    // Each thread owns 4 consecutive x pixels: one NT B128 load per offset
    // plane, one NT B128 store for the output (streams touched exactly once
    // -> NT keeps the 59MB source image resident in the 192MB L2).
    const int xg = (tid & 15) * 4;           // x-group within tile (16B aligned)
    const int x  = tx0 + xg;                 // 1280/64 == 20 tiles: x+3 < W always
    for (int ry = (tid >> 4); ry < TILE; ry += 16) {   // 4 passes, 16 px/thread
        const int y = ty0 + ry;
        if (y >= IMG_H) break;               // bottom partial tile
        const size_t pidx = (size_t)y * IMG_W + x;

        const v4f oy = __builtin_nontemporal_load((const v4f*)&offY[pidx]);
        const v4f ox = __builtin_nontemporal_load((const v4f*)&offX[pidx]);

        v4f r;
        r.x = wgt * sample_px(smem, img, y, x + 0, oy.x, ox.x, gy0, gx0) + bia;
        r.y = wgt * sample_px(smem, img, y, x + 1, oy.y, ox.y, gy0, gx0) + bia;
        r.z = wgt * sample_px(smem, img, y, x + 2, oy.z, ox.z, gy0, gx0) + bia;
        r.w = wgt * sample_px(smem, img, y, x + 3, oy.w, ox.w, gy0, gx0) + bia;

        __builtin_nontemporal_store(r, (v4f*)&outp[pidx]);
    }
}

extern "C" void kernel_launch(void* const* d_in, const int* in_sizes, int n_in,
                              void* d_out, int out_size, void* d_ws, size_t ws_size,
                              hipStream_t stream) {
    (void)in_sizes; (void)n_in; (void)d_ws; (void)ws_size; (void)out_size;
    const float* realB  = (const float*)d_in[0];  // [16,1,720,1280]
    const float* offset = (const float*)d_in[1];  // [16,2,720,1280]
    const float* weight = (const float*)d_in[2];  // [1,1,1,1]
    const float* bias   = (const float*)d_in[3];  // [1]
    float* out = (float*)d_out;                   // [16,1,720,1280]

    dim3 grid((IMG_W + TILE - 1) / TILE,   // 20
              (IMG_H + TILE - 1) / TILE,   // 12
              IMG_B);                      // 16
    blurnet_warp_kernel<<<grid, 256, 0, stream>>>(realB, offset, weight, bias, out);
}